// MultiHotEmbedding_48704929136830
// MI455X (gfx1250) — compile-verified
//
#include <hip/hip_runtime.h>

// Problem constants (match reference)
#define N_CLS    10000
#define D_DIM    64
#define M_TOTAL  3200                       // 16*200 rows
#define M_TILES  200                        // 16-row tiles
#define K_SPLIT  25                         // K chunks (occupancy vs atomic budget)
#define K_CHUNK  (N_CLS / K_SPLIT)          // 400
#define K_HALF   (K_CHUNK / 2)              // 200 rows staged per LDS fill (50 steps)
#define WAVES_PB 4                          // 128-thread blocks; all 4 waves share kc
#define N_BLOCKS (M_TILES / WAVES_PB * K_SPLIT)  // 1250

typedef __attribute__((ext_vector_type(2))) float        v2f;
typedef __attribute__((ext_vector_type(4))) float        v4f;
typedef __attribute__((ext_vector_type(8))) float        v8f;
typedef __attribute__((ext_vector_type(4))) unsigned int v4u;
typedef __attribute__((ext_vector_type(4))) int          v4i;
typedef __attribute__((ext_vector_type(8))) int          v8i;

#if __has_builtin(__builtin_amdgcn_tensor_load_to_lds)
#define USE_TDM 1
#else
#define USE_TDM 0
#endif

static __device__ __forceinline__ void wait_tensorcnt0() {
#if __has_builtin(__builtin_amdgcn_s_wait_tensorcnt)
    __builtin_amdgcn_s_wait_tensorcnt(0);
#else
    asm volatile("s_wait_tensorcnt 0x0" ::: "memory");
#endif
}

#if USE_TDM
// Build a 2D TDM descriptor: tile = K_HALF rows x 64 cols of f32, row stride 64,
// from global `gsrc` into LDS byte offset `lds_off`, and issue TENSOR_LOAD_TO_LDS.
static __device__ __forceinline__ void tdm_stage_b(const float* gsrc, unsigned lds_off) {
    unsigned long long ga = (unsigned long long)(uintptr_t)gsrc;
    v4u g0;
    g0[0] = 1u;                                        // count=1, user mode
    g0[1] = lds_off;                                   // lds_addr (bytes)
    g0[2] = (unsigned)(ga & 0xffffffffu);              // global_addr[31:0]
    g0[3] = (unsigned)((ga >> 32) & 0x01ffffffu)       // global_addr[56:32]
            | (2u << 30);                              // type=2 ("image")
    v8i g1;
    g1[0] = (int)(2u << 16);                           // wg_mask=0, data_size=2 (4B)
    g1[1] = (int)((unsigned)D_DIM << 16);              // tensor_dim0[15:0]=64
    g1[2] = (int)((unsigned)K_HALF << 16);             // tensor_dim1[15:0]=200
    g1[3] = (int)((unsigned)D_DIM << 16);              // tile_dim0=64
    g1[4] = (int)K_HALF;                               // tile_dim1=200, tile_dim2=0
    g1[5] = (int)D_DIM;                                // tensor_dim0_stride[31:0]=64
    g1[6] = 0;                                         // stride hi / dim1_stride lo
    g1[7] = 0;
    v4i z4 = {0, 0, 0, 0};
#if __clang_major__ >= 23
    v8i z8 = {0, 0, 0, 0, 0, 0, 0, 0};
    __builtin_amdgcn_tensor_load_to_lds(g0, g1, z4, z4, z8, 0);
#else
    __builtin_amdgcn_tensor_load_to_lds(g0, g1, z4, z4, 0);
#endif
}
#endif

// ---------------------------------------------------------------------------
// Kernel 0: zero output accumulator (d_out) and row counts (d_ws)
// ---------------------------------------------------------------------------
__global__ void mh_emb_zero(float* __restrict__ out, float* __restrict__ cnt) {
    int i = blockIdx.x * blockDim.x + threadIdx.x;
    if (i < M_TOTAL * D_DIM) out[i] = 0.0f;
    if (i < M_TOTAL)         cnt[i] = 0.0f;
}

// ---------------------------------------------------------------------------
// Kernel 1: fused K-split GEMM + row-count accumulation.
// Block = 4 waves sharing one K chunk; B slab staged in LDS (TDM when available).
// Inner loop software-pipelined: issue next A (NT b64) + next B (ds_load) before
// the WMMAs that consume the current fragments, so no per-step load stall.
// ---------------------------------------------------------------------------
__global__ __launch_bounds__(WAVES_PB * 32)
void mh_emb_gemm(const float* __restrict__ x,
                 const float* __restrict__ w,
                 float* __restrict__ out,
                 float* __restrict__ cnt) {
    __shared__ __align__(16) float ldsB[K_HALF * D_DIM];   // 51200 B

    const int lane = threadIdx.x & 31;
    const int wid  = blockIdx.x * WAVES_PB + (threadIdx.x >> 5);
    const int mt   = wid % M_TILES;          // 16-row tile (consecutive in block)
    const int kc   = wid / M_TILES;          // K chunk (uniform across block)
    const int row0 = mt * 16;
    const int kbase = kc * K_CHUNK;
    const int hi   = lane >> 4;              // 0: K pair {0,1}, 1: K pair {2,3}
    const int l    = lane & 15;

    // A stream: row (row0+l), starting at kbase + 2*hi (8B-aligned float2 steps)
    const float* aPtr = x + (size_t)(row0 + l) * N_CLS + (size_t)kbase + 2 * hi;

    v8f acc0 = {}, acc1 = {}, acc2 = {}, acc3 = {};
    float csum = 0.0f;
    const int base = 2 * hi * D_DIM + l;     // LDS float index at k=0

    for (int h = 0; h < 2; ++h) {
        if (h) __syncthreads();              // readers of previous half done
        const float* wsrc = w + (size_t)(kbase + h * K_HALF) * D_DIM;
#if USE_TDM
        if ((threadIdx.x >> 5) == 0) {       // one wave issues the DMA
            unsigned lds_off =
                (unsigned)(uintptr_t)(__attribute__((address_space(3))) float*)&ldsB[0];
            tdm_stage_b(wsrc, lds_off);
            wait_tensorcnt0();               // TENSORcnt==0 -> slab resident
        }
#else
        {   // cooperative vectorized copy fallback (coalesced b128 both sides)
            const v4f* s4 = (const v4f*)wsrc;
            v4f*       d4 = (v4f*)ldsB;
            for (int i = threadIdx.x; i < K_HALF * D_DIM / 4; i += WAVES_PB * 32)
                d4[i] = s4[i];
        }
#endif
        __syncthreads();

        // ---- software-pipelined inner loop over this half (50 steps of K=4) ----
        v2f a_cur = __builtin_nontemporal_load((const v2f*)aPtr);
        v2f b0c = { ldsB[base +  0], ldsB[base + D_DIM +  0] };
        v2f b1c = { ldsB[base + 16], ldsB[base + D_DIM + 16] };
        v2f b2c = { ldsB[base + 32], ldsB[base + D_DIM + 32] };
        v2f b3c = { ldsB[base + 48], ldsB[base + D_DIM + 48] };

        for (int k = 0; k < K_HALF; k += 4) {
            v2f a_nxt = a_cur, b0n = b0c, b1n = b1c, b2n = b2c, b3n = b3c;
            if (k + 4 < K_HALF) {            // folds under full unroll
                a_nxt = __builtin_nontemporal_load((const v2f*)(aPtr + 4));
                const int nb = base + (k + 4) * D_DIM;
                b0n.x = ldsB[nb +  0]; b0n.y = ldsB[nb + D_DIM +  0];
                b1n.x = ldsB[nb + 16]; b1n.y = ldsB[nb + D_DIM + 16];
                b2n.x = ldsB[nb + 32]; b2n.y = ldsB[nb + D_DIM + 32];
                b3n.x = ldsB[nb + 48]; b3n.y = ldsB[nb + D_DIM + 48];
            }
            if ((k & 127) == 0)              // ~1KB/lane lead into L2, 2x per half
                __builtin_prefetch(aPtr + 256, 0, 1);

            csum += a_cur.x + a_cur.y;       // fused row-count partial
            acc0 = __builtin_amdgcn_wmma_f32_16x16x4_f32(false, a_cur, false, b0c,
                                                         (short)0, acc0, false, false);
            acc1 = __builtin_amdgcn_wmma_f32_16x16x4_f32(false, a_cur, false, b1c,
                                                         (short)0, acc1, false, false);
            acc2 = __builtin_amdgcn_wmma_f32_16x16x4_f32(false, a_cur, false, b2c,
                                                         (short)0, acc2, false, false);
            acc3 = __builtin_amdgcn_wmma_f32_16x16x4_f32(false, a_cur, false, b3c,
                                                         (short)0, acc3, false, false);
            a_cur = a_nxt; b0c = b0n; b1c = b1n; b2c = b2n; b3c = b3n;
            aPtr += 4;
        }
    }

    // Row counts: lanes l and l+16 hold complementary K partials of row (row0+l).
    float ctot = csum + __shfl_xor(csum, 16, 32);
    if (hi == 0) atomicAdd(&cnt[row0 + l], ctot);

    // Tile writeback: acc element r -> row (row0 + 8*hi + r), col (ntile*16 + l)
    float* obase = out + (size_t)(row0 + 8 * hi) * D_DIM + l;
#pragma unroll
    for (int r = 0; r < 8; ++r) {
        atomicAdd(obase + (size_t)r * D_DIM +  0, acc0[r]);
        atomicAdd(obase + (size_t)r * D_DIM + 16, acc1[r]);
        atomicAdd(obase + (size_t)r * D_DIM + 32, acc2[r]);
        atomicAdd(obase + (size_t)r * D_DIM + 48, acc3[r]);
    }
}

// ---------------------------------------------------------------------------
// Kernel 2: out[m][d] /= max(count[m], 1)   (reference's clip(sum, 1, None))
// ---------------------------------------------------------------------------
__global__ void mh_emb_norm(float* __restrict__ out, const float* __restrict__ cnt) {
    int i = blockIdx.x * blockDim.x + threadIdx.x;
    if (i >= M_TOTAL * D_DIM) return;
    float c = cnt[i >> 6];                   // D_DIM == 64
    c = (c < 1.0f) ? 1.0f : c;
    out[i] = out[i] / c;
}

extern "C" void kernel_launch(void* const* d_in, const int* in_sizes, int n_in,
                              void* d_out, int out_size, void* d_ws, size_t ws_size,
                              hipStream_t stream) {
    const float* x = (const float*)d_in[0];  // [3200, 10000] f32
    const float* w = (const float*)d_in[1];  // [10000, 64]   f32
    float* out = (float*)d_out;              // [3200, 64]    f32
    float* cnt = (float*)d_ws;               // [3200]        f32 scratch

    const int total = M_TOTAL * D_DIM;       // 204800
    mh_emb_zero<<<(total + 255) / 256, 256, 0, stream>>>(out, cnt);
    mh_emb_gemm<<<N_BLOCKS, WAVES_PB * 32, 0, stream>>>(x, w, out, cnt);
    mh_emb_norm<<<(total + 255) / 256, 256, 0, stream>>>(out, cnt);
}